// SelfAttention_54425825575468
// MI455X (gfx1250) — compile-verified
//
#include <hip/hip_runtime.h>

// CDNA5 / gfx1250. wave32.
typedef __attribute__((ext_vector_type(2))) float v2f;
typedef __attribute__((ext_vector_type(8))) float v8f;

#define CCH  3
#define HW   96
#define NPIX (HW * HW)   // 9216

// ---------------------------------------------------------------------------
// Kernel 0: zero the per-batch 3x3 accumulators in workspace (B * 9 floats).
// ---------------------------------------------------------------------------
__global__ void zero_ws_kernel(float* __restrict__ Sws, int nelem) {
    int i = blockIdx.x * blockDim.x + threadIdx.x;
    if (i < nelem) Sws[i] = 0.0f;
}

// ---------------------------------------------------------------------------
// Kernel 1: S[b][c][d] = sum over pixels m of V[c,m] * K[d,m]
//   V = Wv @ softmax_C(x) + bv ; K = Wk @ softmax_C(x) + bk
// Each wave accumulates a private 16x16 f32 WMMA accumulator whose top-left
// 3x3 block is its partial S; 4 pixels per v_wmma_f32_16x16x4_f32.
//   A(16x4)[m][k] = V[chan m][pixel p4+k], B(4x16)[k][n] = K[chan n][pixel p4+k]
// f32 A-layout (ISA 7.12.2): lanes 0-15 hold M=lane, vgpr0=K0, vgpr1=K1;
// lanes 16-31 hold vgpr0=K2, vgpr1=K3. B(4x16) stripes rows {0,2}/{1,3} the
// same way, so lane L covers (chan = L&15) at pixels p4 + 2*(L>>4) + {0,1}.
// Wave index is readfirstlane'd so the loop is compiled wave-uniform (SALU
// branch, EXEC untouched) — WMMA's EXEC==all-ones restriction holds by
// construction.
// ---------------------------------------------------------------------------
__global__ __launch_bounds__(256) void kv_outer_kernel(
    const float* __restrict__ x,
    const float* __restrict__ Wk, const float* __restrict__ bk,
    const float* __restrict__ Wv, const float* __restrict__ bv,
    float* __restrict__ Sws)
{
    const int b     = blockIdx.y;
    const int lane  = threadIdx.x & 31;
    const int chan  = lane & 15;
    const int half  = lane >> 4;
    // Scalar (wave-uniform) wave index / stride: keeps the loop under SALU
    // control so EXEC stays all-ones around the WMMAs.
    const int wave  = __builtin_amdgcn_readfirstlane(
                          blockIdx.x * (blockDim.x >> 5) + (threadIdx.x >> 5));
    const int nwav  = __builtin_amdgcn_readfirstlane(
                          gridDim.x * (blockDim.x >> 5));

    const float* xb = x + (size_t)b * CCH * NPIX;

    // This lane's rows of Wk / Wv (zero for padding channels 3..15).
    float wk0 = 0.f, wk1 = 0.f, wk2 = 0.f, bkc = 0.f;
    float wv0 = 0.f, wv1 = 0.f, wv2 = 0.f, bvc = 0.f;
    if (chan < CCH) {
        wk0 = Wk[chan * 3 + 0]; wk1 = Wk[chan * 3 + 1]; wk2 = Wk[chan * 3 + 2];
        wv0 = Wv[chan * 3 + 0]; wv1 = Wv[chan * 3 + 1]; wv2 = Wv[chan * 3 + 2];
        bkc = bk[chan];         bvc = bv[chan];
    }

    v8f acc = {};
    const int ngroups = NPIX / 4;                 // 2304
    for (int g = wave; g < ngroups; g += nwav) {  // uniform loop
        const int p4 = g * 4;
        const int pa = p4 + half * 2;             // this lane's 2 pixels
        float av[2], bv2[2];
#pragma unroll
        for (int j = 0; j < 2; ++j) {
            const int p = pa + j;
            const float x0 = xb[0 * NPIX + p];
            const float x1 = xb[1 * NPIX + p];
            const float x2 = xb[2 * NPIX + p];
            const float mx = fmaxf(x0, fmaxf(x1, x2));
            const float e0 = __expf(x0 - mx), e1 = __expf(x1 - mx), e2 = __expf(x2 - mx);
            const float inv = 1.0f / (e0 + e1 + e2);
            const float s0 = e0 * inv, s1 = e1 * inv, s2 = e2 * inv;
            const float kq = wk0 * s0 + wk1 * s1 + wk2 * s2 + bkc;
            const float vq = wv0 * s0 + wv1 * s1 + wv2 * s2 + bvc;
            av[j]  = (chan < CCH) ? vq : 0.0f;    // A = V
            bv2[j] = (chan < CCH) ? kq : 0.0f;    // B = K
        }
        v2f a;  a[0]  = av[0];  a[1]  = av[1];
        v2f bb; bb[0] = bv2[0]; bb[1] = bv2[1];
        acc = __builtin_amdgcn_wmma_f32_16x16x4_f32(
            false, a, false, bb, (short)0, acc, false, false);
    }

    // D[r][n], r<3, lanes 0-2 carry S[r][lane]; flush partials to global S.
    if (lane < CCH) {
        float* Sb = Sws + b * 9;
        __hip_atomic_fetch_add(&Sb[0 * 3 + lane], acc[0],
                               __ATOMIC_RELAXED, __HIP_MEMORY_SCOPE_AGENT);
        __hip_atomic_fetch_add(&Sb[1 * 3 + lane], acc[1],
                               __ATOMIC_RELAXED, __HIP_MEMORY_SCOPE_AGENT);
        __hip_atomic_fetch_add(&Sb[2 * 3 + lane], acc[2],
                               __ATOMIC_RELAXED, __HIP_MEMORY_SCOPE_AGENT);
    }
}

// ---------------------------------------------------------------------------
// Kernel 2: fuse M = Wp@S@Wq (3x3), v = Wp@S@bq + bp (3), then per pixel
//   out[:,p] = M @ softmax_C(x[:,p]) + v + x[:,p]
// One v_wmma_f32_16x16x4_f32 applies M to 16 pixels:
//   A(16x4)[o][e] = M[o][e] (loop-invariant), B(4x16)[e][n] = xn[e][g16+n]
// ---------------------------------------------------------------------------
__global__ __launch_bounds__(256) void attn_apply_kernel(
    const float* __restrict__ x,
    const float* __restrict__ Wq, const float* __restrict__ bq,
    const float* __restrict__ Wp, const float* __restrict__ bp,
    const float* __restrict__ Sws,
    float* __restrict__ out)
{
    const int b    = blockIdx.y;
    const int lane = threadIdx.x & 31;
    const int chan = lane & 15;
    const int half = lane >> 4;
    const int wave = __builtin_amdgcn_readfirstlane(
                         blockIdx.x * (blockDim.x >> 5) + (threadIdx.x >> 5));
    const int nwav = __builtin_amdgcn_readfirstlane(
                         gridDim.x * (blockDim.x >> 5));

    // Fused channel map (each thread redundantly; 3x3 algebra, L2-resident).
    const float* Sb = Sws + b * 9;
    float S[9], U[9], Mm[9], u[3], vb[3];
#pragma unroll
    for (int i = 0; i < 9; ++i) S[i] = Sb[i];
#pragma unroll
    for (int c = 0; c < 3; ++c) {
#pragma unroll
        for (int e = 0; e < 3; ++e)
            U[c * 3 + e] = S[c * 3 + 0] * Wq[0 * 3 + e]
                         + S[c * 3 + 1] * Wq[1 * 3 + e]
                         + S[c * 3 + 2] * Wq[2 * 3 + e];
        u[c] = S[c * 3 + 0] * bq[0] + S[c * 3 + 1] * bq[1] + S[c * 3 + 2] * bq[2];
    }
#pragma unroll
    for (int o = 0; o < 3; ++o) {
#pragma unroll
        for (int e = 0; e < 3; ++e)
            Mm[o * 3 + e] = Wp[o * 3 + 0] * U[0 * 3 + e]
                          + Wp[o * 3 + 1] * U[1 * 3 + e]
                          + Wp[o * 3 + 2] * U[2 * 3 + e];
        vb[o] = Wp[o * 3 + 0] * u[0] + Wp[o * 3 + 1] * u[1]
              + Wp[o * 3 + 2] * u[2] + bp[o];
    }

    // Loop-invariant A operand = M (rows 0-2 of 16, K-cols 0-2 of 4).
    v2f a; a[0] = 0.f; a[1] = 0.f;
    if (chan < CCH) {
        if (half == 0) { a[0] = Mm[chan * 3 + 0]; a[1] = Mm[chan * 3 + 1]; }
        else           { a[0] = Mm[chan * 3 + 2]; /* K=3 column is zero */ }
    }

    const float* xb = x   + (size_t)b * CCH * NPIX;
    float*       ob = out + (size_t)b * CCH * NPIX;

    const int ngroups = NPIX / 16;                // 576
    for (int g = wave; g < ngroups; g += nwav) {  // uniform loop
        const int pix = g * 16 + chan;            // both halves: same 16 pixels
        const float x0 = xb[0 * NPIX + pix];
        const float x1 = xb[1 * NPIX + pix];
        const float x2 = xb[2 * NPIX + pix];
        const float mx = fmaxf(x0, fmaxf(x1, x2));
        const float e0 = __expf(x0 - mx), e1 = __expf(x1 - mx), e2 = __expf(x2 - mx);
        const float inv = 1.0f / (e0 + e1 + e2);

        // B rows {0,2} in comp0, rows {1,3} in comp1 (row = input channel).
        v2f bb;
        bb[0] = half ? (e2 * inv) : (e0 * inv);
        bb[1] = half ? 0.0f       : (e1 * inv);

        v8f acc = {};
        acc = __builtin_amdgcn_wmma_f32_16x16x4_f32(
            false, a, false, bb, (short)0, acc, false, false);

        // D row r, lanes 0-15 = out-channel r at pixel g16+lane.
        if (half == 0) {
            ob[0 * NPIX + pix] = acc[0] + vb[0] + x0;
            ob[1 * NPIX + pix] = acc[1] + vb[1] + x1;
            ob[2 * NPIX + pix] = acc[2] + vb[2] + x2;
        }
    }
}

// ---------------------------------------------------------------------------
// Host launcher (graph-capture safe: only kernel launches on `stream`).
// Inputs: x, Wq, bq, Wk, bk, Wv, bv, Wp, bp  (all float32)
// ---------------------------------------------------------------------------
extern "C" void kernel_launch(void* const* d_in, const int* in_sizes, int n_in,
                              void* d_out, int out_size, void* d_ws, size_t ws_size,
                              hipStream_t stream) {
    const float* x  = (const float*)d_in[0];
    const float* Wq = (const float*)d_in[1];
    const float* bq = (const float*)d_in[2];
    const float* Wk = (const float*)d_in[3];
    const float* bk = (const float*)d_in[4];
    const float* Wv = (const float*)d_in[5];
    const float* bv = (const float*)d_in[6];
    const float* Wp = (const float*)d_in[7];
    const float* bp = (const float*)d_in[8];
    float* out = (float*)d_out;
    float* Sws = (float*)d_ws;       // B * 9 floats of S = V K^T partial sums

    const int B = in_sizes[0] / (CCH * NPIX);   // = 2

    zero_ws_kernel<<<1, 64, 0, stream>>>(Sws, B * 9);
    kv_outer_kernel<<<dim3(8, B), 256, 0, stream>>>(x, Wk, bk, Wv, bv, Sws);
    attn_apply_kernel<<<dim3(8, B), 256, 0, stream>>>(x, Wq, bq, Wp, bp, Sws, out);
}